// InceptionBlock_71631464562989
// MI455X (gfx1250) — compile-verified
//
#include <hip/hip_runtime.h>

typedef __attribute__((ext_vector_type(2))) float v2f;
typedef __attribute__((ext_vector_type(8))) float v8f;

#define NNODES 10000
#define FDIM   256
#define MTILES (NNODES / 16)   // 625
#define NTILES (FDIM / 16)     // 16

// ---------------------------------------------------------------------------
// Dense GEMM: H[N,256] = X[N,256] @ W[256,256], fp32 WMMA 16x16x4.
// One wave computes one 16x16 tile of H. Grid = 625*16 = 10000 waves exactly,
// so EXEC is all-ones for every wave (WMMA requirement).
//
// A fragment (16x4 fp32, ISA 7.12.2): lanes 0-15 hold K=k+0 (v0), k+1 (v1)
// for row M=lane; lanes 16-31 hold K=k+2, k+3 for row M=lane-16.
// B fragment (4x16): v0 = row k+half*2, v1 = row k+half*2+1, col = lane&15.
// C/D (16x16 f32, 8 VGPRs): VGPR i -> row i + 8*(lane>=16), col = lane&15.
// ---------------------------------------------------------------------------
__global__ __launch_bounds__(256) void gemm_wmma_f32(const float* __restrict__ X,
                                                     const float* __restrict__ W,
                                                     float* __restrict__ H) {
  const int wave = (int)((blockIdx.x * blockDim.x + threadIdx.x) >> 5);
  const int lane = (int)(threadIdx.x & 31);
  const int tm = wave / NTILES;      // 0..624
  const int tn = wave % NTILES;      // 0..15
  if (tm >= MTILES) return;          // never taken (wave-uniform), keeps EXEC full

  const int half = lane >> 4;        // 0: K={0,1}, 1: K={2,3}
  const int l15  = lane & 15;

  const float* ap = X + (size_t)(tm * 16 + l15) * FDIM + half * 2;   // row-major A
  const float* bp = W + (size_t)(half * 2) * FDIM + tn * 16 + l15;   // row-major B

  v8f acc = {};
  for (int k = 0; k < FDIM; k += 4) {
    v2f a;
    a.x = ap[k];
    a.y = ap[k + 1];
    v2f b;
    b.x = bp[(size_t)k * FDIM];
    b.y = bp[(size_t)(k + 1) * FDIM];
    // (neg_a, A, neg_b, B, c_mod, C, reuse_a, reuse_b)
    acc = __builtin_amdgcn_wmma_f32_16x16x4_f32(false, a, false, b,
                                                (short)0, acc, false, false);
  }

  float* op = H + (size_t)(tm * 16 + half * 8) * FDIM + tn * 16 + l15;
#pragma unroll
  for (int i = 0; i < 8; ++i) op[(size_t)i * FDIM] = acc[i];
}

// ---------------------------------------------------------------------------
// Scatter SpMM: OUT[dst[e], :] += vals[e] * H[src[e], :]
// One wave per edge -> vals/src/dst are wave-uniform (scalar loads).
// Each lane handles 8 contiguous floats (two float4 gathers), 8 fp32 HW
// atomics (global_atomic_add_f32 via unsafeAtomicAdd; L2-resident traffic).
// ---------------------------------------------------------------------------
__global__ __launch_bounds__(256) void spmm_scatter(const float* __restrict__ vals,
                                                    const int* __restrict__ src,
                                                    const int* __restrict__ dst,
                                                    const float* __restrict__ H,
                                                    float* __restrict__ OUT,
                                                    int nedges) {
  const int e = (int)(blockIdx.x * 8 + (threadIdx.x >> 5));
  if (e >= nedges) return;
  const int lane = (int)(threadIdx.x & 31);

  const float v = vals[e];
  const int s = src[e];
  const int d = dst[e];

  const float* h = H + (size_t)s * FDIM + lane * 8;
  float* o = OUT + (size_t)d * FDIM + lane * 8;

  float4 h0 = *(const float4*)(h);
  float4 h1 = *(const float4*)(h + 4);

  unsafeAtomicAdd(o + 0, v * h0.x);
  unsafeAtomicAdd(o + 1, v * h0.y);
  unsafeAtomicAdd(o + 2, v * h0.z);
  unsafeAtomicAdd(o + 3, v * h0.w);
  unsafeAtomicAdd(o + 4, v * h1.x);
  unsafeAtomicAdd(o + 5, v * h1.y);
  unsafeAtomicAdd(o + 6, v * h1.z);
  unsafeAtomicAdd(o + 7, v * h1.w);
}

// ---------------------------------------------------------------------------
// out = relu(out / 3), vectorized float4, in place.
// ---------------------------------------------------------------------------
__global__ __launch_bounds__(256) void relu_scale(float* __restrict__ out, int n4) {
  const int i = (int)(blockIdx.x * blockDim.x + threadIdx.x);
  if (i >= n4) return;
  const float s = 1.0f / 3.0f;
  float4 v = ((float4*)out)[i];
  v.x = fmaxf(v.x * s, 0.0f);
  v.y = fmaxf(v.y * s, 0.0f);
  v.z = fmaxf(v.z * s, 0.0f);
  v.w = fmaxf(v.w * s, 0.0f);
  ((float4*)out)[i] = v;
}

extern "C" void kernel_launch(void* const* d_in, const int* in_sizes, int n_in,
                              void* d_out, int out_size, void* d_ws, size_t ws_size,
                              hipStream_t stream) {
  const float* x      = (const float*)d_in[0];
  const float* A_vals = (const float*)d_in[1];
  const float* W1     = (const float*)d_in[2];
  const float* W2     = (const float*)d_in[3];
  const float* W3     = (const float*)d_in[4];
  const int*   esrc   = (const int*)d_in[5];
  const int*   edst   = (const int*)d_in[6];
  float* out = (float*)d_out;
  const int E = in_sizes[1];

  const size_t nf = (size_t)NNODES * FDIM;
  float* bufA = (float*)d_ws;        // N*F fp32
  float* bufB = bufA + nf;           // N*F fp32  (total ws use: 20.5 MB)

  const dim3 blk(256);
  const dim3 gemmGrid((MTILES * NTILES) / 8);        // 1250 blocks, 8 waves each
  const dim3 spmmGrid((unsigned)((E + 7) / 8));
  const dim3 reluGrid((unsigned)((nf / 4 + 255) / 256));

  // acc = d_out, must start at zero (harness poisons it).
  hipMemsetAsync(out, 0, nf * sizeof(float), stream);

  // Branch 1: f1 = A @ (x W1)
  gemm_wmma_f32<<<gemmGrid, blk, 0, stream>>>(x, W1, bufA);
  spmm_scatter<<<spmmGrid, blk, 0, stream>>>(A_vals, esrc, edst, bufA, out, E);

  // Branch 2: f2 = A @ (A @ (x W2))
  gemm_wmma_f32<<<gemmGrid, blk, 0, stream>>>(x, W2, bufA);
  hipMemsetAsync(bufB, 0, nf * sizeof(float), stream);
  spmm_scatter<<<spmmGrid, blk, 0, stream>>>(A_vals, esrc, edst, bufA, bufB, E);
  spmm_scatter<<<spmmGrid, blk, 0, stream>>>(A_vals, esrc, edst, bufB, out, E);

  // Branch 3: f3 = A @ (A @ (A @ (x W3)))
  gemm_wmma_f32<<<gemmGrid, blk, 0, stream>>>(x, W3, bufA);
  hipMemsetAsync(bufB, 0, nf * sizeof(float), stream);
  spmm_scatter<<<spmmGrid, blk, 0, stream>>>(A_vals, esrc, edst, bufA, bufB, E);
  hipMemsetAsync(bufA, 0, nf * sizeof(float), stream);
  spmm_scatter<<<spmmGrid, blk, 0, stream>>>(A_vals, esrc, edst, bufB, bufA, E);
  spmm_scatter<<<spmmGrid, blk, 0, stream>>>(A_vals, esrc, edst, bufA, out, E);

  // out = relu((f1+f2+f3)/3)
  relu_scale<<<reluGrid, blk, 0, stream>>>(out, (int)(nf / 4));
}